// TemporalVortexController_18691697672684
// MI455X (gfx1250) — compile-verified
//
#include <hip/hip_runtime.h>
#include <hip/hip_bf16.h>
#include <stdint.h>

#define T_STEPS 8192
#define N_NODES 4096
#define TT      256     // time columns per block (= blockDim.x)
#define RSLAB   4       // rows per LDS slab
#define NBUF    3       // pipeline depth (2-slab prefetch lead)
#define PI_F    3.14159265358979323846f
#define TWO_PI_F 6.28318530717958647692f

// Low 32 bits of a generic pointer to __shared__ data == LDS byte offset
// (CDNA5 ISA 10.2: LDS aperture addresses truncate to addr[31:0]).
__device__ __forceinline__ uint32_t lds_addr_of(const void* p) {
  return (uint32_t)(uintptr_t)p;
}

// GVS-mode async copy: copies from the global base in saddr plus the offset in vaddr to the LDS address in vdst, 16 bytes per lane.
// Tracked by ASYNCcnt; loads complete in order per wave. Data is streamed
// exactly once (256 MB > 192 MB L2), so use the non-temporal hint.
__device__ __forceinline__ void async_load_b128(uint32_t lds_byte_addr,
                                                int gbyte_off,
                                                uint64_t base) {
  asm volatile("global_load_async_to_lds_b128 %0, %1, %2 th:TH_LOAD_NT"
               :: "v"(lds_byte_addr), "v"(gbyte_off), "s"(base)
               : "memory");
}

__device__ __forceinline__ float wrap_pi(float d) {
  d = (d >  PI_F) ? d - TWO_PI_F : d;
  d = (d < -PI_F) ? d + TWO_PI_F : d;
  return d;
}

__global__ __launch_bounds__(TT)
void vortex_partial(const float* __restrict__ re,
                    const float* __restrict__ im,
                    float* __restrict__ ws_mag,
                    float* __restrict__ ws_wind,
                    int rows_per_chunk) {
  __shared__ float s_re[NBUF][RSLAB * TT];   // 12 KB
  __shared__ float s_im[NBUF][RSLAB * TT];   // 12 KB

  const int tid = threadIdx.x;                 // 0..255
  const int t0  = blockIdx.x * TT;             // time tile start
  const int n0  = blockIdx.y * rows_per_chunk; // row chunk start
  const int nslab = rows_per_chunk / RSLAB;    // >= 64

  // loader mapping: 256 threads cover a 4-row x 256-col slab, 16B per lane
  const int row_l = tid >> 6;          // 0..3
  const int cg    = (tid & 63) << 2;   // 0,4,...,252

  const uint64_t re_base = (uint64_t)(uintptr_t)re;
  const uint64_t im_base = (uint64_t)(uintptr_t)im;

  auto issue = [&](int s, int b) {
    int gb = (((n0 + s * RSLAB + row_l) * T_STEPS) + t0 + cg) * 4;
    async_load_b128(lds_addr_of(&s_re[b][row_l * TT + cg]), gb, re_base);
    async_load_b128(lds_addr_of(&s_im[b][row_l * TT + cg]), gb, im_base);
  };

  float mag_sum  = 0.0f;
  float wind_sum = 0.0f;
  float prev_ph  = 0.0f;

  // prologue: 2 slabs in flight
  issue(0, 0);
  issue(1, 1);

  int buf = 0;
  for (int s = 0; s < nslab; ++s) {
    // keep a 2-slab prefetch lead
    if (s + 2 < nslab) {
      int bnext = buf + 2;
      bnext = (bnext >= NBUF) ? bnext - NBUF : bnext;
      issue(s + 2, bnext);
    }
    // async loads complete in order: leaving 2 per outstanding future slab
    // guarantees slab s has fully landed in LDS (own wave's share).
    if (s + 3 <= nslab) {
      asm volatile("s_wait_asynccnt 0x4" ::: "memory");
    } else if (s + 2 == nslab) {
      asm volatile("s_wait_asynccnt 0x2" ::: "memory");
    } else {
      asm volatile("s_wait_asynccnt 0x0" ::: "memory");
    }
    __syncthreads();                             // all waves' copies visible
#pragma unroll
    for (int r = 0; r < RSLAB; ++r) {
      float vr = s_re[buf][r * TT + tid];
      float vi = s_im[buf][r * TT + tid];
      mag_sum += __builtin_sqrtf(vr * vr + vi * vi);
      float ph = atan2f(vi, vr);
      if (!(s == 0 && r == 0)) {
        wind_sum += wrap_pi(ph - prev_ph);
      }
      prev_ph = ph;
    }
    __syncthreads();       // all reads of buf done before slab s+3 reuses it
    buf = (buf + 1 >= NBUF) ? 0 : buf + 1;
  }

  // boundary diff with first row of the next chunk (direct load)
  const int nlast = n0 + rows_per_chunk;
  const int t = t0 + tid;
  if (nlast < N_NODES) {
    float vr = re[(size_t)nlast * T_STEPS + t];
    float vi = im[(size_t)nlast * T_STEPS + t];
    float ph = atan2f(vi, vr);
    wind_sum += wrap_pi(ph - prev_ph);
  }

  ws_mag[(size_t)blockIdx.y * T_STEPS + t]  = mag_sum;
  ws_wind[(size_t)blockIdx.y * T_STEPS + t] = wind_sum;
}

__global__ __launch_bounds__(256)
void vortex_final(const float* __restrict__ ws_mag,
                  const float* __restrict__ ws_wind,
                  float* __restrict__ out, int nchunk) {
  const int t = blockIdx.x * blockDim.x + threadIdx.x;
  if (t >= T_STEPS) return;
  float m = 0.0f, w = 0.0f;
  for (int c = 0; c < nchunk; ++c) {
    m += ws_mag[(size_t)c * T_STEPS + t];
    w += ws_wind[(size_t)c * T_STEPS + t];
  }
  const float mean = m * (1.0f / (float)N_NODES);
  const float wind = w * (1.0f / TWO_PI_F);
  const bool vortex = (mean < 2.0f) && (fabsf(wind) > 0.5f);
  out[t]               = mean;
  out[T_STEPS + t]     = vortex ? wind : 0.0f;
  out[2 * T_STEPS + t] = vortex ? 1.0f : 0.0f;
}

extern "C" void kernel_launch(void* const* d_in, const int* in_sizes, int n_in,
                              void* d_out, int out_size, void* d_ws, size_t ws_size,
                              hipStream_t stream) {
  const float* re = (const float*)d_in[0];
  const float* im = (const float*)d_in[1];
  float* out = (float*)d_out;

  // Deterministic: nchunk depends only on the fixed ws_size.
  int nchunk = 16;
  while (nchunk > 1 &&
         (size_t)2 * (size_t)nchunk * T_STEPS * sizeof(float) > ws_size) {
    nchunk >>= 1;
  }
  const int rows = N_NODES / nchunk;

  float* ws_mag  = (float*)d_ws;
  float* ws_wind = ws_mag + (size_t)nchunk * T_STEPS;

  dim3 grid(T_STEPS / TT, nchunk);
  vortex_partial<<<grid, TT, 0, stream>>>(re, im, ws_mag, ws_wind, rows);
  vortex_final<<<T_STEPS / 256, 256, 0, stream>>>(ws_mag, ws_wind, out, nchunk);
}